// FM_89429809037693
// MI455X (gfx1250) — compile-verified
//
#include <hip/hip_runtime.h>

typedef __attribute__((ext_vector_type(2))) float v2f;
typedef __attribute__((ext_vector_type(8))) float v8f;
typedef int v4i __attribute__((vector_size(16)));   // matches async builtin param type

#define FDIM 128       // feature dim = 2*D
#define KDIM 16        // FM rank
#define DDIM 64        // embedding dim
#define XPAD 132       // padded LDS row stride (floats): 132%64=4 -> conflict-free WMMA A loads
#define SPB  64        // samples per block
#define THREADS 128    // 4 waves

// single-instruction xor-shuffle within wave32: ds_swizzle group-of-32 mode
#define SWZ_XOR(x, M) \
    __int_as_float(__builtin_amdgcn_ds_swizzle(__float_as_int(x), (((M) << 10) | 0x1f)))

#if __has_builtin(__builtin_amdgcn_global_load_async_to_lds_b128)
#define HAVE_ASYNC 1
#else
#define HAVE_ASYNC 0
#endif

#if HAVE_ASYNC
__device__ __forceinline__ void async_copy16(const void* g, void* l) {
    // memory -> LDS, 16B per lane, tracked by ASYNCcnt (no VGPR data staging)
    __builtin_amdgcn_global_load_async_to_lds_b128((v4i*)g, (v4i*)l,
                                                   /*offset=*/0, /*cpol=*/0);
}
__device__ __forceinline__ void async_wait0() {
#if __has_builtin(__builtin_amdgcn_s_wait_asynccnt)
    __builtin_amdgcn_s_wait_asynccnt(0);
#else
    asm volatile("s_wait_asynccnt 0x0" ::: "memory");
#endif
}
#endif

__global__ __launch_bounds__(THREADS)
void fm_fwd_kernel(const int* __restrict__ user,
                   const int* __restrict__ item,
                   const float* __restrict__ user_emb,
                   const float* __restrict__ item_emb,
                   const float* __restrict__ w1,
                   const float* __restrict__ b1,
                   const float* __restrict__ v,
                   float* __restrict__ out,
                   int B)
{
    __shared__ float xs[SPB * XPAD];       // gathered x tile, padded rows
    __shared__ float vs[FDIM * KDIM];      // v, row-major [128][16]
    __shared__ float w1s[FDIM];            // w1
    __shared__ float red[4][16];           // per-wave pair-term scratch

    const int tid  = threadIdx.x;
    const int lane = tid & 31;
    const int wave = tid >> 5;
    const int blockBase = blockIdx.x * SPB;

    // ---- Gather x tile into LDS. Per wave-iteration: one full 128-float row
    //      (contiguous 256B user half + 256B item half). Loop-invariant parts
    //      (table select, column) hoisted: lane q is the float4 column.
    {
        const int q    = lane;                          // 0..31 float4 column
        const int scol = (q < 16) ? q * 4 : (q - 16) * 4;
        const float* ebase = (q < 16) ? user_emb : item_emb;
        const int*   iarr  = (q < 16) ? user : item;
        #pragma unroll 4
        for (int j = 0; j < 16; ++j) {
            int s   = wave + 4 * j;                     // sample within block
            int gs  = blockBase + s;
            int cgs = gs < B ? gs : B - 1;              // clamp tail (store guarded later)
            const float* src = ebase + (size_t)iarr[cgs] * DDIM + scol;
            float* dst = &xs[s * XPAD + q * 4];         // 16B aligned (XPAD*4 = 33*16)
#if HAVE_ASYNC
            async_copy16(src, dst);
#else
            *(float4*)dst = *(const float4*)src;
#endif
        }
    }
    // ---- Stage v (8KB)
    for (int j = 0; j < 4; ++j) {
        int fidx = tid + THREADS * j;                   // 0..511 float4s
#if HAVE_ASYNC
        async_copy16((const float4*)v + fidx, (float4*)vs + fidx);
#else
        ((float4*)vs)[fidx] = ((const float4*)v)[fidx];
#endif
    }
    if (tid < FDIM) w1s[tid] = w1[tid];
#if HAVE_ASYNC
    async_wait0();                                      // LDS writes landed before barrier
#endif
    __syncthreads();

    // ---- Dual WMMA chains (fp32, K-chunks of 4):
    //      D1 = x@v ;  D2 = (x*x)@(v*v)  (A2/B2 are elementwise squares of the
    //      already-loaded fragments -- zero extra LDS traffic)
    const int m  = lane & 15;                 // row of A / column of B
    const int kh = (lane >> 4) << 1;          // K sub-offset: 0 (lanes 0-15) or 2 (lanes 16-31)
    const float* xrow = &xs[(wave * 16 + m) * XPAD];
    v8f acc1 = {0.f, 0.f, 0.f, 0.f, 0.f, 0.f, 0.f, 0.f};
    v8f acc2 = {0.f, 0.f, 0.f, 0.f, 0.f, 0.f, 0.f, 0.f};

    #pragma unroll 4
    for (int t = 0; t < FDIM / 4; ++t) {
        int k0 = t * 4 + kh;
        v2f a = { xrow[k0], xrow[k0 + 1] };                      // A: VGPR0=K even, VGPR1=K odd
        v2f b = { vs[k0 * KDIM + m], vs[(k0 + 1) * KDIM + m] };  // B: row k0/k0+1, col m
        v2f a2 = a * a;
        v2f b2 = b * b;
        acc1 = __builtin_amdgcn_wmma_f32_16x16x4_f32(false, a,  false, b,  (short)0, acc1, false, false);
        acc2 = __builtin_amdgcn_wmma_f32_16x16x4_f32(false, a2, false, b2, (short)0, acc2, false, false);
    }

    // ---- pair term: reduce rows of (D1^2 - D2) across N (16 lanes per half)
    //      acc[r]: lanes 0-15 = row r, lanes 16-31 = row r+8
    float sel = 0.f;
    #pragma unroll
    for (int r = 0; r < 8; ++r) {
        float y = acc1[r] * acc1[r] - acc2[r];
        y += SWZ_XOR(y, 8);
        y += SWZ_XOR(y, 4);
        y += SWZ_XOR(y, 2);
        y += SWZ_XOR(y, 1);
        sel = ((lane & 15) == r) ? y : sel;   // v_cndmask select, no exec juggling
    }
    if ((lane & 15) < 8)
        red[wave][(lane >> 4) * 8 + (lane & 15)] = sel;  // sample r (lanes<16) / r+8 (lanes>=16)

    // ---- linear term: lin_m = x . w1  (one 64-feature half per 16-lane group)
    float linp = 0.f;
    {
        const int h = lane >> 4;
        const float* xr2 = &xs[(wave * 16 + m) * XPAD + h * 64];
        #pragma unroll 4
        for (int f = 0; f < 64; ++f)
            linp += xr2[f] * w1s[h * 64 + f];
    }
    linp += SWZ_XOR(linp, 16);

    // ---- finalize (same-wave DS ops are in-order; red[] written above by this wave)
    if (lane < 16) {
        int gs = blockBase + wave * 16 + lane;
        if (gs < B)
            out[gs] = linp + b1[0] + 0.5f * red[wave][lane];
    }
}

extern "C" void kernel_launch(void* const* d_in, const int* in_sizes, int n_in,
                              void* d_out, int out_size, void* d_ws, size_t ws_size,
                              hipStream_t stream) {
    const int*   user     = (const int*)d_in[0];
    const int*   item     = (const int*)d_in[1];
    const float* user_emb = (const float*)d_in[2];
    const float* item_emb = (const float*)d_in[3];
    const float* w1       = (const float*)d_in[4];
    const float* b1       = (const float*)d_in[5];
    const float* v        = (const float*)d_in[6];
    float*       out      = (float*)d_out;

    int B = in_sizes[0];
    int grid = (B + SPB - 1) / SPB;
    fm_fwd_kernel<<<grid, THREADS, 0, stream>>>(user, item, user_emb, item_emb,
                                                w1, b1, v, out, B);
}